// DummyMoveNet_47845935677587
// MI455X (gfx1250) — compile-verified
//
#include <hip/hip_runtime.h>
#include <hip/hip_bf16.h>

// MoveNet-style pose decode for MI455X (gfx1250, wave32).
//
// One workgroup per batch element; TDM async-loads hm (17x24x24) + ct (24x24)
// into LDS (tensor_load_to_lds, TENSORcnt); bilinear 2x upsample expressed as
// Y = R_h * X * R_w^T and executed with V_WMMA_F32_16X16X4_F32; the GEMM2
// epilogue fuses the distance-weighted argmax so the 48x48 resized maps never
// leave registers. rg/of are only gathered at single pixels (4-tap footprint).

#define NJ 17
#define HIN 24
#define WIN 24
#define HT 48
#define WT 48
#define NPIX (HT * WT)              // 2304
#define CH_ELEMS (HIN * WIN)        // 576
#define HM_ELEMS (NJ * CH_ELEMS)    // 9792

typedef float v2f __attribute__((ext_vector_type(2)));
typedef float v8f __attribute__((ext_vector_type(8)));
typedef unsigned int u32;
typedef u32 u32x4 __attribute__((ext_vector_type(4)));
typedef int i32x4 __attribute__((ext_vector_type(4)));
typedef int i32x8 __attribute__((ext_vector_type(8)));

#if __has_builtin(__builtin_amdgcn_tensor_load_to_lds)
#define HAVE_TDM 1
#else
#define HAVE_TDM 0
#endif

// Generic pointer -> LDS byte offset (addrspacecast to AS(3), 32-bit).
__device__ __forceinline__ unsigned ldsByteOffset(void* p) {
  return (unsigned)(unsigned long long)(__attribute__((address_space(3))) void*)p;
}

#if HAVE_TDM
// 1-D contiguous DMA of nElems f32 from global to LDS via the Tensor Data
// Mover. D# layout per CDNA5 ISA:
//   group0: count=1 | lds_addr | global_addr[56:0] | type=2
//   group1: data_size=4B, tensor_dim0=tile_dim0=nElems, dims1..=1, stride0=nElems
__device__ __forceinline__ void tdm_load_1d(unsigned ldsOff, const void* gptr,
                                            unsigned nElems) {
  unsigned long long ga = (unsigned long long)gptr;
  u32x4 g0;
  g0.x = 1u;                                            // count=1 (user mode)
  g0.y = ldsOff;                                        // lds_addr (bytes)
  g0.z = (u32)ga;                                       // global_addr[31:0]
  g0.w = (u32)((ga >> 32) & 0x01ffffffull) | (2u << 30);// addr[56:32] | type=2
  i32x8 g1;
  g1[0] = (int)(2u << 16);                              // data_size=4B, mask=0
  g1[1] = (int)((nElems & 0xffffu) << 16);              // tensor_dim0[15:0]
  g1[2] = (int)(((nElems >> 16) & 0xffffu) | (1u << 16)); // dim0[31:16]|dim1=1
  g1[3] = (int)((nElems & 0xffffu) << 16);              // tile_dim0
  g1[4] = 1;                                            // tile_dim1=1, dim2=0
  g1[5] = (int)nElems;                                  // tensor_dim0_stride
  g1[6] = 0;
  g1[7] = 0;
  i32x4 gz = {0, 0, 0, 0};
#if __clang_major__ >= 23
  i32x8 gz8 = {0, 0, 0, 0, 0, 0, 0, 0};
  __builtin_amdgcn_tensor_load_to_lds(g0, g1, gz, gz, gz8, 0);
#else
  __builtin_amdgcn_tensor_load_to_lds(g0, g1, gz, gz, 0);
#endif
}
#endif

// align_corners=False 2x upsample taps: output o samples input at o/2 - 0.25.
// Returns the two source indices (clamped) and the weight of i1.
__device__ __forceinline__ void taps(int o, int& i0, int& i1, float& f1) {
  int k = o >> 1;
  if (o & 1) { i0 = k; i1 = (k + 1 < HIN) ? k + 1 : HIN - 1; f1 = 0.25f; }
  else       { i0 = (k > 0) ? k - 1 : 0; i1 = k;             f1 = 0.75f; }
}

__device__ __forceinline__ float tap4(const float* p, int y0, int y1, float fy,
                                      int x0, int x1, float fx) {
  float a = p[y0 * WIN + x0], b = p[y0 * WIN + x1];
  float c = p[y1 * WIN + x0], d = p[y1 * WIN + x1];
  return (1.f - fy) * ((1.f - fx) * a + fx * b) +
         fy * ((1.f - fx) * c + fx * d);
}

// One 16x16 f32 WMMA output tile of A(MxK) * B(KxN), K multiple of 4.
// Fragment layouts per CDNA5 ISA:
//   A 16x4: lanes 0-15 rows m0+l, VGPR0/1 = K {0,1}; lanes 16-31 = K {2,3}
//   B 4x16: lanes 0-15 cols n0+l, VGPR0/1 = K {0,1}; lanes 16-31 = K {2,3}
//   C/D:    VGPR r = row m0+r (lanes 0-15) / m0+8+r (lanes 16-31), col n0+l
//
// Out-of-range A rows are CLAMPED, not zeroed: in 16x16x4 WMMA, D row m
// depends only on A row m, and the clamped rows' D values are discarded by
// the caller. This keeps the inner loop free of EXEC-mask manipulation.
__device__ __forceinline__ v8f wmma_tile(const float* A, int lda, int m0,
                                         const float* B, int ldb, int n0,
                                         int K, int mLimit, int hlf, int l16) {
  int row = m0 + l16;
  if (row >= mLimit) row = mLimit - 1;   // branch-free clamp (uniform-safe)
  const float* Arow = A + row * lda;
  const float* Bcol = B + n0 + l16;
  v8f acc = {};
  for (int k0 = 0; k0 < K; k0 += 4) {
    int ka = k0 + 2 * hlf;
    v2f af, bf;
    af.x = Arow[ka];
    af.y = Arow[ka + 1];
    bf.x = Bcol[ka * ldb];
    bf.y = Bcol[(ka + 1) * ldb];
    acc = __builtin_amdgcn_wmma_f32_16x16x4_f32(false, af, false, bf,
                                                (short)0, acc, false, false);
  }
  return acc;
}

__global__ __launch_bounds__(256) void movenet_decode(
    const float* __restrict__ hm, const float* __restrict__ ct,
    const float* __restrict__ rg, const float* __restrict__ of,
    float* __restrict__ out) {
  __shared__ float sHM[HM_ELEMS];      // 17 heatmap channels, 24x24
  __shared__ float sCT[CH_ELEMS];      // center channel, 24x24
  __shared__ float sAh[HT * HIN];      // R_h (48x24) row-major
  __shared__ float sBw[HIN * WT];      // R_w^T (24x48) row-major
  __shared__ float sT[HIN * WT];       // GEMM1 result T (24x48)
  __shared__ float sRegs[2 * NJ];
  __shared__ float sRegX[NJ];
  __shared__ float sRegY[NJ];
  __shared__ int   sCenter[2];         // cy, cx
  __shared__ float sRedT[8];
  __shared__ float sRedV[8];
  __shared__ int   sRedI[8];
  __shared__ int   sJidx[NJ];
  __shared__ float sJscore[NJ];

  const int tid  = threadIdx.x;
  const int wave = tid >> 5;
  const int lane = tid & 31;
  const int hlf  = lane >> 4;
  const int l16  = lane & 15;
  const int b    = blockIdx.x;

  const float* hmB = hm + (size_t)b * HM_ELEMS;
  const float* ctB = ct + (size_t)b * CH_ELEMS;

#if HAVE_TDM
  if (wave == 0) {  // wave-uniform branch: one wave issues the DMAs
    tdm_load_1d(ldsByteOffset(sHM), hmB, HM_ELEMS);
    tdm_load_1d(ldsByteOffset(sCT), ctB, CH_ELEMS);
  }
#endif

  // Warm GL2 for the later single-pixel rg/of gathers.
  if (tid < 2 * NJ) {
    __builtin_prefetch(rg + ((size_t)b * 2 * NJ + tid) * CH_ELEMS, 0, 0);
    __builtin_prefetch(of + ((size_t)b * 2 * NJ + tid) * CH_ELEMS, 0, 0);
  }

  // Build interpolation matrices while the DMA is in flight.
  for (int e = tid; e < HT * HIN; e += 256) {
    int o = e / HIN, i = e - o * HIN;
    int i0, i1; float f1;
    taps(o, i0, i1, f1);
    float w = ((i == i0) ? (1.f - f1) : 0.f) + ((i == i1) ? f1 : 0.f);
    sAh[o * HIN + i] = w;
    sBw[i * WT + o] = w;
  }

#if HAVE_TDM
  if (wave == 0) {
#if __has_builtin(__builtin_amdgcn_s_wait_tensorcnt)
    __builtin_amdgcn_s_wait_tensorcnt(0);
#endif
  }
#else
  for (int i = tid; i < HM_ELEMS; i += 256) sHM[i] = hmB[i];
  for (int i = tid; i < CH_ELEMS; i += 256) sCT[i] = ctB[i];
#endif
  __syncthreads();

  // ---- Phase 1: center argmax over resized ct (scalar 4-tap, 9 px/thread)
  {
    float bt = -3.4e38f; int bi = NPIX;
    for (int p = tid; p < NPIX; p += 256) {
      int y = p / WT, x = p - y * WT;
      int y0, y1, x0, x1; float fy, fx;
      taps(y, y0, y1, fy); taps(x, x0, x1, fx);
      float v = tap4(sCT, y0, y1, fy, x0, x1, fx);
      if (v > bt || (v == bt && p < bi)) { bt = v; bi = p; }
    }
    for (int off = 16; off; off >>= 1) {
      float ot = __shfl_xor(bt, off);
      int oi = __shfl_xor(bi, off);
      if (ot > bt || (ot == bt && oi < bi)) { bt = ot; bi = oi; }
    }
    if (lane == 0) { sRedT[wave] = bt; sRedI[wave] = bi; }
    __syncthreads();
    if (tid == 0) {
      float ft = sRedT[0]; int fi = sRedI[0];
      for (int w = 1; w < 8; ++w)
        if (sRedT[w] > ft || (sRedT[w] == ft && sRedI[w] < fi)) {
          ft = sRedT[w]; fi = sRedI[w];
        }
      sCenter[0] = fi / WT;   // cy
      sCenter[1] = fi % WT;   // cx
    }
    __syncthreads();
  }

  // ---- Phase 2: gather rg at center (4-tap footprint), build reg_x/reg_y
  if (tid < 2 * NJ) {
    int cy = sCenter[0], cx = sCenter[1];
    int y0, y1, x0, x1; float fy, fx;
    taps(cy, y0, y1, fy); taps(cx, x0, x1, fx);
    const float* rgC = rg + ((size_t)b * 2 * NJ + tid) * CH_ELEMS;
    sRegs[tid] = tap4(rgC, y0, y1, fy, x0, x1, fx);
  }
  __syncthreads();
  if (tid < NJ) {
    int cy = sCenter[0], cx = sCenter[1];
    float rx = (float)cx + sRegs[2 * tid] + 0.5f;
    float ry = (float)cy + sRegs[2 * tid + 1] + 0.5f;
    sRegX[tid] = fminf(fmaxf(rx, 0.f), (float)(WT - 1));
    sRegY[tid] = fminf(fmaxf(ry, 0.f), (float)(HT - 1));
  }
  __syncthreads();

  // ---- Phase 3: per joint, WMMA resize fused with distance-weighted argmax
  for (int j = 0; j < NJ; ++j) {
    const float* X = sHM + j * CH_ELEMS;

    // GEMM1: T(24x48) = X(24x24) @ Bw(24x48); 2x3 tiles on waves 0..5.
    if (wave < 6) {
      int m0 = (wave / 3) * 16;
      int n0 = (wave % 3) * 16;
      v8f acc = wmma_tile(X, WIN, m0, sBw, WT, n0, 24, HIN, hlf, l16);
      for (int r = 0; r < 8; ++r) {
        int row = m0 + r + 8 * hlf;
        if (row < HIN) sT[row * WT + n0 + l16] = acc[r];
      }
    }
    __syncthreads();

    // GEMM2: Y(48x48) = Ah(48x24) @ T(24x48); 9 tiles over 8 waves,
    // epilogue computes tmp = Y * rsqrt(d2 + 1e-9) / 1.8 and argmaxes.
    float regx = sRegX[j], regy = sRegY[j];
    float bT = -3.4e38f, bV = 0.f; int bI = NPIX;
    for (int t = wave; t < 9; t += 8) {
      int m0 = (t / 3) * 16;
      int n0 = (t % 3) * 16;
      v8f acc = wmma_tile(sAh, HIN, m0, sT, WT, n0, 24, HT, hlf, l16);
      int x = n0 + l16;
      float dx = (float)x - regx;
      float dx2 = dx * dx;
      for (int r = 0; r < 8; ++r) {
        int y = m0 + r + 8 * hlf;
        float dy = (float)y - regy;
        float val = acc[r];
        float tmp = val * rsqrtf(dx2 + dy * dy + 1e-9f) * (1.0f / 1.8f);
        int idx = y * WT + x;
        if (tmp > bT || (tmp == bT && idx < bI)) { bT = tmp; bV = val; bI = idx; }
      }
    }
    for (int off = 16; off; off >>= 1) {
      float ot = __shfl_xor(bT, off);
      float ov = __shfl_xor(bV, off);
      int oi = __shfl_xor(bI, off);
      if (ot > bT || (ot == bT && oi < bI)) { bT = ot; bV = ov; bI = oi; }
    }
    if (lane == 0) { sRedT[wave] = bT; sRedV[wave] = bV; sRedI[wave] = bI; }
    __syncthreads();
    if (tid == 0) {
      float fT = sRedT[0], fV = sRedV[0]; int fI = sRedI[0];
      for (int w = 1; w < 8; ++w)
        if (sRedT[w] > fT || (sRedT[w] == fT && sRedI[w] < fI)) {
          fT = sRedT[w]; fV = sRedV[w]; fI = sRedI[w];
        }
      sJidx[j] = fI; sJscore[j] = fV;
    }
    __syncthreads();
  }

  // ---- Phase 4: gather offsets at joint pixels, emit (x, y, score) per joint
  if (tid < NJ) {
    int j = tid;
    int jidx = sJidx[j];
    int jy = jidx / WT, jx = jidx - jy * WT;
    int y0, y1, x0, x1; float fy, fx;
    taps(jy, y0, y1, fy); taps(jx, x0, x1, fx);
    const float* ofx = of + ((size_t)b * 2 * NJ + 2 * j) * CH_ELEMS;
    const float* ofy = ofx + CH_ELEMS;
    float offx = tap4(ofx, y0, y1, fy, x0, x1, fx);
    float offy = tap4(ofy, y0, y1, fy, x0, x1, fx);
    float* o = out + (size_t)b * 3 * NJ + 3 * j;
    o[0] = ((float)jx + offx) / (float)WT;
    o[1] = ((float)jy + offy) / (float)HT;
    o[2] = sJscore[j];
  }
}

extern "C" void kernel_launch(void* const* d_in, const int* in_sizes, int n_in,
                              void* d_out, int out_size, void* d_ws,
                              size_t ws_size, hipStream_t stream) {
  const float* hm = (const float*)d_in[0];
  const float* ct = (const float*)d_in[1];
  const float* rg = (const float*)d_in[2];
  const float* of = (const float*)d_in[3];
  float* out = (float*)d_out;
  int B = in_sizes[0] / HM_ELEMS;   // 1024
  movenet_decode<<<dim3(B), dim3(256), 0, stream>>>(hm, ct, rg, of, out);
}